// EstimationGate_6640019440014
// MI455X (gfx1250) — compile-verified
//
#include <hip/hip_runtime.h>
#include <hip/hip_bf16.h>

typedef __attribute__((ext_vector_type(2))) float v2f;
typedef __attribute__((ext_vector_type(4))) float f4;
typedef __attribute__((ext_vector_type(8))) float v8f;

#define N_NODES 2048
#define E_DIM   64
#define H_DIM   64

// ---------------------------------------------------------------------------
// Kernel 1: per-node gate MLP using V_WMMA_F32_16X16X4_F32 (exact fp32).
// One wave (32 threads) computes a 16-node tile:
//   hidden[16,64] = relu(concat(u,d)[16,128] @ W1[128,64] + b1)
//   gate[16]      = sigmoid(hidden @ W2[64,1] + b2)
// ---------------------------------------------------------------------------
__global__ __launch_bounds__(32) void gate_mlp_wmma(
    const float* __restrict__ emb_u,   // [N, E]
    const float* __restrict__ emb_d,   // [N, E]
    const float* __restrict__ W1,      // [2E, H]
    const float* __restrict__ b1,      // [H]
    const float* __restrict__ W2,      // [H, 1]
    const float* __restrict__ b2,      // [1]
    float* __restrict__ gate)          // [N]
{
    const int lane  = threadIdx.x & 31;
    const int half  = lane >> 4;       // 0: K={0,1}, rows 0-7 of C ; 1: K={2,3}, rows 8-15
    const int l16   = lane & 15;
    const int node0 = blockIdx.x << 4; // 16 nodes per wave
    const int arow  = node0 + l16;     // A-matrix row this lane feeds

    v8f acc[4] = {};                   // 4 N-tiles of the 16x64 hidden matrix

    // K = 0..63 sourced from emb_u
    #pragma unroll 4
    for (int kk = 0; kk < 16; ++kk) {
        const int k0 = 4 * kk + 2 * half;           // this lane's first K
        v2f A = { emb_u[arow * E_DIM + k0], emb_u[arow * E_DIM + k0 + 1] };
        #pragma unroll
        for (int n = 0; n < 4; ++n) {
            const int col = n * 16 + l16;           // B column this lane feeds
            v2f Bv = { W1[k0 * H_DIM + col], W1[(k0 + 1) * H_DIM + col] };
            acc[n] = __builtin_amdgcn_wmma_f32_16x16x4_f32(
                false, A, false, Bv, (short)0, acc[n], false, false);
        }
    }
    // K = 64..127 sourced from emb_d
    #pragma unroll 4
    for (int kk = 0; kk < 16; ++kk) {
        const int kg = 64 + 4 * kk + 2 * half;      // global K index into W1
        v2f A = { emb_d[arow * E_DIM + (kg - 64)], emb_d[arow * E_DIM + (kg - 63)] };
        #pragma unroll
        for (int n = 0; n < 4; ++n) {
            const int col = n * 16 + l16;
            v2f Bv = { W1[kg * H_DIM + col], W1[(kg + 1) * H_DIM + col] };
            acc[n] = __builtin_amdgcn_wmma_f32_16x16x4_f32(
                false, A, false, Bv, (short)0, acc[n], false, false);
        }
    }

    // Epilogue: +b1, relu, dot with W2 along the lane's 4 columns.
    // C/D layout: VGPR r, lanes 0-15 -> (M=r, N=l16); lanes 16-31 -> (M=8+r, N=l16).
    float res[8];
    #pragma unroll
    for (int r = 0; r < 8; ++r) res[r] = 0.f;
    #pragma unroll
    for (int n = 0; n < 4; ++n) {
        const int   col  = n * 16 + l16;
        const float bias = b1[col];
        const float w2   = W2[col];
        #pragma unroll
        for (int r = 0; r < 8; ++r) {
            float h = acc[n][r] + bias;
            h = fmaxf(h, 0.f);
            res[r] += h * w2;
        }
    }
    // Reduce across the 16 lanes of each half (xor masks <= 8 stay in-half).
    #pragma unroll
    for (int off = 8; off >= 1; off >>= 1) {
        #pragma unroll
        for (int r = 0; r < 8; ++r)
            res[r] += __shfl_xor(res[r], off, 32);
    }

    if (l16 == 0) {
        const float bias2 = b2[0];
        #pragma unroll
        for (int r = 0; r < 8; ++r) {
            const float x = res[r] + bias2;
            gate[node0 + half * 8 + r] = 1.f / (1.f + __expf(-x));
        }
    }
}

// ---------------------------------------------------------------------------
// Kernel 2: streaming broadcast multiply. One float4 (B128) per thread,
// non-temporal so the 805 MB stream bypasses cache retention while the
// 8 KB gate table stays hot.
// ---------------------------------------------------------------------------
__global__ __launch_bounds__(256) void gate_scale(
    const f4* __restrict__ hist,
    const float* __restrict__ gate,
    f4* __restrict__ out,
    long long total4)
{
    const long long idx = (long long)blockIdx.x * blockDim.x + threadIdx.x;
    if (idx >= total4) return;
    // C = 32 floats = 8 float4 per node row; node index cycles every 8 vectors.
    const int n = (int)((idx >> 3) & (N_NODES - 1));
    const float g = gate[n];
    f4 v = __builtin_nontemporal_load(&hist[idx]);
    v *= g;
    __builtin_nontemporal_store(v, &out[idx]);
}

extern "C" void kernel_launch(void* const* d_in, const int* in_sizes, int n_in,
                              void* d_out, int out_size, void* d_ws, size_t ws_size,
                              hipStream_t stream) {
    const float* emb_u = (const float*)d_in[0];
    const float* emb_d = (const float*)d_in[1];
    const float* hist  = (const float*)d_in[2];
    const float* W1    = (const float*)d_in[3];
    const float* b1    = (const float*)d_in[4];
    const float* W2    = (const float*)d_in[5];
    const float* b2    = (const float*)d_in[6];
    float* out  = (float*)d_out;
    float* gate = (float*)d_ws;            // 2048 floats of scratch

    // 2048 nodes / 16 per wave = 128 single-wave blocks.
    gate_mlp_wmma<<<N_NODES / 16, 32, 0, stream>>>(emb_u, emb_d, W1, b1, W2, b2, gate);

    const long long total4 = (long long)out_size / 4;   // float4 count
    const int threads = 256;
    const long long blocks = (total4 + threads - 1) / threads;
    gate_scale<<<(unsigned int)blocks, threads, 0, stream>>>(
        (const f4*)hist, gate, (f4*)out, total4);
}